// ContrastiveLoss_43834436223453
// MI455X (gfx1250) — compile-verified
//
#include <hip/hip_runtime.h>

// ContrastiveLoss on MI455X (gfx1250, wave32).
// Bandwidth-bound: 134 MB single-pass stream -> ~5.8 us floor at 23.3 TB/s.
// Kernel 1: per-row sum of squared differences, b128 coalesced loads,
//           wave reduction via V_WMMA_F32_16X16X4_F32 (f32-exact).
// Kernel 2: argmax over 22 classes, contrastive select, final scalar reduce.

#define IGNORE_INDEX 21

typedef __attribute__((ext_vector_type(2))) float v2f;
typedef __attribute__((ext_vector_type(8))) float v8f;

// Full 32-lane sum, result valid in every lane.
// A-matrix 16x4 f32 layout (ISA 7.12.2): lane L, VGPR0 -> (M=L%16, K = 2*(L/16)),
// VGPR1 -> K+1. Put partial in VGPR0, 0 in VGPR1; B = ones(4x16).
// D[m,n] = p[m] + p[m+16] (replicated over n). Lane n holds D[M=8*(n/16)+r, *]
// in its 8 C/D VGPRs, so summing them + shfl_xor(16) yields the full wave sum.
__device__ __forceinline__ float wave_sum_wmma(float partial) {
  v2f a; a[0] = partial; a[1] = 0.0f;
  v2f b; b[0] = 1.0f;    b[1] = 1.0f;
  v8f c = {};
  c = __builtin_amdgcn_wmma_f32_16x16x4_f32(
      /*neg_a=*/false, a, /*neg_b=*/false, b,
      /*c_mod=*/(short)0, c, /*reuse_a=*/false, /*reuse_b=*/false);
  float s = ((c[0] + c[1]) + (c[2] + c[3])) + ((c[4] + c[5]) + (c[6] + c[7]));
  s += __shfl_xor(s, 16, 32);
  return s;
}

__device__ __forceinline__ int argmax22(const float* __restrict__ p) {
  float best = p[0];
  int bi = 0;
#pragma unroll
  for (int c = 1; c < 22; ++c) {
    float v = p[c];
    if (v > best) { best = v; bi = c; }  // strict '>' keeps first max (jnp.argmax)
  }
  return bi;
}

// One block per (b,s) row. 256 threads x 4 float4 each = 4096 floats per tensor row.
__global__ __launch_bounds__(256) void cl_row_sqdist(const float4* __restrict__ x,
                                                     const float4* __restrict__ y,
                                                     float* __restrict__ d) {
  const int VEC_PER_ROW = 1024;  // 4096 floats / 4
  const long base = (long)blockIdx.x * VEC_PER_ROW;
  float acc = 0.0f;
#pragma unroll
  for (int j = 0; j < 4; ++j) {
    int i = threadIdx.x + j * 256;           // uniform trip count, no divergence
    float4 a = x[base + i];
    float4 b = y[base + i];
    float dx = a.x - b.x, dy = a.y - b.y, dz = a.z - b.z, dw = a.w - b.w;
    acc = fmaf(dx, dx, acc);
    acc = fmaf(dy, dy, acc);
    acc = fmaf(dz, dz, acc);
    acc = fmaf(dw, dw, acc);
  }

  __shared__ float wsum[8];
  float s = wave_sum_wmma(acc);              // EXEC all-1s here
  const int lane = threadIdx.x & 31;
  const int wv   = threadIdx.x >> 5;
  if (lane == 0) wsum[wv] = s;
  __syncthreads();
  if (threadIdx.x == 0) {
    float t = 0.0f;
#pragma unroll
    for (int i = 0; i < 8; ++i) t += wsum[i];
    d[blockIdx.x] = t;
  }
}

// Single block: classes + contrastive terms + final reduction (0.36 MB of reads).
__global__ __launch_bounds__(256) void cl_finalize(const float* __restrict__ yts,
                                                   const float* __restrict__ d,
                                                   const int* __restrict__ m_ptr,
                                                   float* __restrict__ out) {
  const int B = 64, S = 64, C = 22, R = B * S;
  __shared__ int   cls0_s[64];
  __shared__ float wsum[8];

  if (threadIdx.x < B) {
    // class of last token of batch b: row b*S + (S-1)
    cls0_s[threadIdx.x] = argmax22(yts + ((long)(threadIdx.x * S + (S - 1))) * C);
  }
  __syncthreads();

  const float mf = (float)(*m_ptr);
  float acc = 0.0f;
#pragma unroll 4
  for (int r = threadIdx.x; r < R; r += 256) {  // 16 uniform iterations
    const int b  = r >> 6;
    const int ct = argmax22(yts + (long)r * C);
    const int c0 = cls0_s[b];
    const bool valid = (ct != IGNORE_INDEX) && (c0 != IGNORE_INDEX);
    const float dv   = d[r];
    const float per  = (ct == c0) ? dv : fmaxf(mf - dv, 0.0f);
    acc += valid ? per : 0.0f;
  }

  float s = wave_sum_wmma(acc);              // EXEC all-1s here
  const int lane = threadIdx.x & 31;
  const int wv   = threadIdx.x >> 5;
  if (lane == 0) wsum[wv] = s;
  __syncthreads();
  if (threadIdx.x == 0) {
    float t = 0.0f;
#pragma unroll
    for (int i = 0; i < 8; ++i) t += wsum[i];
    out[0] = t / (float)R;
  }
}

extern "C" void kernel_launch(void* const* d_in, const int* in_sizes, int n_in,
                              void* d_out, int out_size, void* d_ws, size_t ws_size,
                              hipStream_t stream) {
  const float4* xtes = (const float4*)d_in[0];  // (64,64,4096) f32
  const float4* x0es = (const float4*)d_in[1];  // (64,64,4096) f32
  const float*  yts  = (const float*)d_in[2];   // (64,64,22)  f32
  const int*    m    = (const int*)d_in[3];     // scalar int
  float* out = (float*)d_out;                   // scalar f32 loss
  float* dws = (float*)d_ws;                    // 4096 f32 row distances

  const int rows = 64 * 64;
  cl_row_sqdist<<<rows, 256, 0, stream>>>(xtes, x0es, dws);
  cl_finalize<<<1, 256, 0, stream>>>(yts, dws, m, out);
}